// SsmTransformerDecoderLayer_68015102099606
// MI455X (gfx1250) — compile-verified
//
#include <hip/hip_runtime.h>
#include <hip/hip_bf16.h>
#include <math.h>

typedef __attribute__((ext_vector_type(2))) float v2f;
typedef __attribute__((ext_vector_type(8))) float v8f;

#define DM   256
#define NH   8
#define DI   512
#define PD   64
#define NDIM 16
#define NQ   300
#define BSZ  2
#define LSEQ 2048
#define DKEY 1034   // 2*DI + 2 + 8

// =====================================================================
// WMMA f32 GEMM (NT): C[m,n] = sum_k A[m,k] * B[n,k]
// A:[M,K] row-major, B:[N,K] row-major (weights), C:[M,ldc].
// Block = 8 waves (2 x 4); wave tile 16x64 = four 16x16 WMMA tiles
// sharing one A fragment. Block tile 32(M) x 256(N).
// Software-pipelined with unroll suppressed so the fragment rotation
// survives scheduling: next k-step's 5 loads stay in flight across the
// 4 independent v_wmma_f32_16x16x4_f32 of the current step.
// =====================================================================
__global__ __launch_bounds__(256) void gemm_nt_f32_wmma(
    const float* __restrict__ A, const float* __restrict__ B,
    float* __restrict__ C, int M, int N, int K, int ldc)
{
  const int lane = threadIdx.x & 31;
  const int wave = threadIdx.x >> 5;      // 0..7
  const int wm   = wave >> 2;             // 0..1
  const int wn   = wave & 3;              // 0..3
  const int blockM = blockIdx.y * 32;
  const int blockN = blockIdx.x * 256;
  const int hi  = lane >> 4;
  const int l15 = lane & 15;

  int mrow = blockM + wm * 16 + l15;
  int mcl  = mrow < M ? mrow : M - 1;
  int nb   = blockN + wn * 64 + l15;
  int n0 = nb, n1 = nb + 16, n2 = nb + 32, n3 = nb + 48;
  int n0c = n0 < N ? n0 : N - 1;
  int n1c = n1 < N ? n1 : N - 1;
  int n2c = n2 < N ? n2 : N - 1;
  int n3c = n3 < N ? n3 : N - 1;

  const int koff = hi * 2;
  v8f acc0 = {0.f,0.f,0.f,0.f,0.f,0.f,0.f,0.f};
  v8f acc1 = acc0, acc2 = acc0, acc3 = acc0;

  const float* Arow = A + (size_t)mcl * K + koff;
  const float* B0   = B + (size_t)n0c * K + koff;
  const float* B1   = B + (size_t)n1c * K + koff;
  const float* B2   = B + (size_t)n2c * K + koff;
  const float* B3   = B + (size_t)n3c * K + koff;

  // prologue: fragments for k = 0
  v2f a  = *(const v2f*)(Arow);
  v2f b0 = *(const v2f*)(B0);
  v2f b1 = *(const v2f*)(B1);
  v2f b2 = *(const v2f*)(B2);
  v2f b3 = *(const v2f*)(B3);

#pragma unroll 1
  for (int k = 4; k < K; k += 4) {
    // issue next step's loads interleaved with current step's WMMAs
    v2f an  = *(const v2f*)(Arow + k);
    v2f b0n = *(const v2f*)(B0 + k);
    acc0 = __builtin_amdgcn_wmma_f32_16x16x4_f32(false, a, false, b0,
                                                 (short)0, acc0, false, false);
    v2f b1n = *(const v2f*)(B1 + k);
    acc1 = __builtin_amdgcn_wmma_f32_16x16x4_f32(false, a, false, b1,
                                                 (short)0, acc1, false, false);
    v2f b2n = *(const v2f*)(B2 + k);
    acc2 = __builtin_amdgcn_wmma_f32_16x16x4_f32(false, a, false, b2,
                                                 (short)0, acc2, false, false);
    v2f b3n = *(const v2f*)(B3 + k);
    acc3 = __builtin_amdgcn_wmma_f32_16x16x4_f32(false, a, false, b3,
                                                 (short)0, acc3, false, false);
    a = an; b0 = b0n; b1 = b1n; b2 = b2n; b3 = b3n;
  }
  // epilogue
  acc0 = __builtin_amdgcn_wmma_f32_16x16x4_f32(false, a, false, b0,
                                               (short)0, acc0, false, false);
  acc1 = __builtin_amdgcn_wmma_f32_16x16x4_f32(false, a, false, b1,
                                               (short)0, acc1, false, false);
  acc2 = __builtin_amdgcn_wmma_f32_16x16x4_f32(false, a, false, b2,
                                               (short)0, acc2, false, false);
  acc3 = __builtin_amdgcn_wmma_f32_16x16x4_f32(false, a, false, b3,
                                               (short)0, acc3, false, false);

  // D layout: VGPR r -> row (base + hi*8 + r), col = l15 within tile
  int rbase = blockM + wm * 16 + hi * 8;
#pragma unroll
  for (int r = 0; r < 8; ++r) {
    int row = rbase + r;
    if (row < M) {
      float* crow = C + (size_t)row * ldc;
      if (n0 < N) crow[n0] = acc0[r];
      if (n1 < N) crow[n1] = acc1[r];
      if (n2 < N) crow[n2] = acc2[r];
      if (n3 < N) crow[n3] = acc3[r];
    }
  }
}

// =====================================================================
// dist projections: one thread per (b,l,q). Reads 16 floats of dist,
// produces dA=exp(dt*A), dtB=dt*Bm (layout [b,l,h,q]) and C ([b,l,q]).
// Weights (W_bc 2x16, W_dt 8x16, dt_bias, -exp(A_log)) staged in LDS.
// =====================================================================
__global__ __launch_bounds__(256) void dist_proj_kernel(
    const float* __restrict__ dist, const float* __restrict__ zx,
    const float* __restrict__ W_bc, const float* __restrict__ W_dt,
    const float* __restrict__ dt_bias, const float* __restrict__ A_log,
    float* __restrict__ dAw, float* __restrict__ dtBw, float* __restrict__ Cw)
{
  __shared__ float sWbc[32], sWdt[128], sB[8], sNA[8];
  int t = threadIdx.x;
  if (t < 32)  sWbc[t] = W_bc[t];
  if (t < 128) sWdt[t] = W_dt[t];
  if (t < 8)   { sB[t] = dt_bias[t]; sNA[t] = -expf(A_log[t]); }
  __syncthreads();

  size_t idx = (size_t)blockIdx.x * 256 + t;
  if (idx >= (size_t)BSZ * LSEQ * NQ) return;
  int q = (int)(idx % NQ);
  size_t bl = idx / NQ;

  const float4* dp = (const float4*)(dist + idx * NDIM);
  float4 v0 = dp[0], v1 = dp[1], v2 = dp[2], v3 = dp[3];
  float d[16] = {v0.x,v0.y,v0.z,v0.w, v1.x,v1.y,v1.z,v1.w,
                 v2.x,v2.y,v2.z,v2.w, v3.x,v3.y,v3.z,v3.w};

  float bc0 = 0.f, bc1 = 0.f;
#pragma unroll
  for (int i = 0; i < 16; ++i) {
    bc0 = fmaf(d[i], sWbc[i], bc0);
    bc1 = fmaf(d[i], sWbc[16 + i], bc1);
  }
  const float* zrow = zx + bl * DKEY;
  float Bv = bc0 + zrow[2 * DI];        // b_bias
  float Cv = bc1 + zrow[2 * DI + 1];    // c_bias
  Cw[bl * NQ + q] = Cv;

#pragma unroll
  for (int h = 0; h < NH; ++h) {
    float s = 0.f;
#pragma unroll
    for (int i = 0; i < 16; ++i) s = fmaf(d[i], sWdt[h * 16 + i], s);
    s += zrow[2 * DI + 2 + h] + sB[h];
    float dtv = s > 20.f ? s : log1pf(expf(s));     // softplus
    size_t o = (bl * NH + h) * NQ + q;
    dAw[o]  = expf(dtv * sNA[h]);
    dtBw[o] = dtv * Bv;
  }
}

// =====================================================================
// Selective scan: 32 blocks = (dir, b, h). State S[64][300] lives in
// registers: thread (p=tid>>2, j=tid&3) owns n in [75j, 75j+75).
// Per step: stage dA/dtB/C rows + x into LDS, 75 unrolled FMAs on the
// register state, y reduced over the 4 n-slices via __shfl_xor (wave32).
// =====================================================================
__global__ __launch_bounds__(256) void scan_kernel(
    const float* __restrict__ zx, const float* __restrict__ dAw,
    const float* __restrict__ dtBw, const float* __restrict__ Cw,
    const float* __restrict__ initf, float* __restrict__ y_ws,
    float* __restrict__ S_ws)
{
  const int blk = blockIdx.x;           // 0..31
  const int dir = blk >> 4;
  const int b   = (blk >> 3) & 1;
  const int h   = blk & 7;
  const int tid = threadIdx.x;
  const int p   = tid >> 2;
  const int j   = tid & 3;
  const int nbase = j * 75;

  __shared__ float sx[PD];
  __shared__ float sA[NQ], sU[NQ], sC[NQ];

  float S[75];
#pragma unroll
  for (int i = 0; i < 75; ++i)          // init[b,q,(h,p)] from init GEMM
    S[i] = initf[(size_t)(b * NQ + nbase + i) * DI + h * PD + p];

  const int d4 = dir * 2 + b;
  for (int l = 0; l < LSEQ; ++l) {
    int tt = dir ? (LSEQ - 1 - l) : l;
    size_t bl = (size_t)b * LSEQ + tt;
    const float* rA = dAw  + (bl * NH + h) * NQ;
    const float* rU = dtBw + (bl * NH + h) * NQ;
    const float* rC = Cw   + bl * NQ;
    for (int s = tid; s < NQ; s += 256) { sA[s] = rA[s]; sU[s] = rU[s]; sC[s] = rC[s]; }
    if (tid < PD) sx[tid] = zx[bl * DKEY + DI + h * PD + tid];
    __syncthreads();

    // prefetch next step's operand rows (global_prefetch_b8)
    if (l + 1 < LSEQ && tid < 8) {
      int t2 = dir ? tt - 1 : tt + 1;
      size_t bl2 = (size_t)b * LSEQ + t2;
      int off = tid * 40;
      __builtin_prefetch(dAw  + (bl2 * NH + h) * NQ + off, 0, 0);
      __builtin_prefetch(dtBw + (bl2 * NH + h) * NQ + off, 0, 0);
      __builtin_prefetch(Cw   + bl2 * NQ + off, 0, 0);
    }

    float xv = sx[p];
    float y = 0.f;
#pragma unroll
    for (int i = 0; i < 75; ++i) {
      int n = nbase + i;
      S[i] = fmaf(S[i], sA[n], sU[n] * xv);
      y = fmaf(S[i], sC[n], y);
    }
    y += __shfl_xor(y, 1, 32);
    y += __shfl_xor(y, 2, 32);
    if (j == 0) y_ws[((size_t)d4 * LSEQ + tt) * DI + h * PD + p] = y;
    __syncthreads();
  }
#pragma unroll
  for (int i = 0; i < 75; ++i)
    S_ws[(((size_t)d4 * NH + h) * PD + p) * NQ + nbase + i] = S[i];
}

// =====================================================================
// y combine + D-skip + SiLU gate + RMSNorm -> kf rows
// =====================================================================
__global__ __launch_bounds__(256) void gate_rms_kernel(
    const float* __restrict__ y_ws, const float* __restrict__ zx,
    const float* __restrict__ Dp, const float* __restrict__ key_norm_w,
    float* __restrict__ kf)
{
  int row = blockIdx.x;                  // b*LSEQ + t
  int b = row >> 11;
  int t = row & (LSEQ - 1);
  __shared__ float gbuf[DI];
  __shared__ float red[256];
  const float* zrow = zx + (size_t)row * DKEY;
  const float* yf = y_ws + ((size_t)b * LSEQ + t) * DI;
  const float* yb = y_ws + ((size_t)(2 + b) * LSEQ + t) * DI;

  float ss = 0.f;
  for (int c = threadIdx.x; c < DI; c += 256) {
    float yv = 0.5f * (yf[c] + yb[c]) + zrow[DI + c] * Dp[c >> 6];
    float zv = zrow[c];
    float g  = yv * (zv / (1.f + expf(-zv)));   // y * silu(z)
    gbuf[c] = g;
    ss = fmaf(g, g, ss);
  }
  red[threadIdx.x] = ss;
  __syncthreads();
  for (int s = 128; s > 0; s >>= 1) {
    if (threadIdx.x < s) red[threadIdx.x] += red[threadIdx.x + s];
    __syncthreads();
  }
  float inv = rsqrtf(red[0] * (1.f / DI) + 1e-5f);
  for (int c = threadIdx.x; c < DI; c += 256)
    kf[(size_t)row * DI + c] = gbuf[c] * inv * key_norm_w[c];
}

// =====================================================================
// final-state combine + LayerNorm -> lastn rows [b*NQ+q, DI]
// =====================================================================
__global__ __launch_bounds__(256) void state_ln_kernel(
    const float* __restrict__ S_ws, const float* __restrict__ qw,
    const float* __restrict__ qb, float* __restrict__ lastn)
{
  int row = blockIdx.x;                  // 0..BSZ*NQ-1
  int b = row / NQ;
  int q = row % NQ;
  __shared__ float buf[DI];
  __shared__ float red1[256], red2[256];
  float s1 = 0.f, s2 = 0.f;
  for (int c = threadIdx.x; c < DI; c += 256) {
    int h = c >> 6, p = c & 63;
    size_t o0 = (((size_t)b * NH + h) * PD + p) * NQ + q;
    size_t o1 = (((size_t)(2 + b) * NH + h) * PD + p) * NQ + q;
    float v = 0.5f * (S_ws[o0] + S_ws[o1]);
    buf[c] = v;
    s1 += v;
    s2 = fmaf(v, v, s2);
  }
  red1[threadIdx.x] = s1; red2[threadIdx.x] = s2;
  __syncthreads();
  for (int s = 128; s > 0; s >>= 1) {
    if (threadIdx.x < s) {
      red1[threadIdx.x] += red1[threadIdx.x + s];
      red2[threadIdx.x] += red2[threadIdx.x + s];
    }
    __syncthreads();
  }
  float mu  = red1[0] * (1.f / DI);
  float var = red2[0] * (1.f / DI) - mu * mu;
  float inv = rsqrtf(var + 1e-5f);
  for (int c = threadIdx.x; c < DI; c += 256)
    lastn[(size_t)row * DI + c] = (buf[c] - mu) * inv * qw[c] + qb[c];
}

// =====================================================================
extern "C" void kernel_launch(void* const* d_in, const int* in_sizes, int n_in,
                              void* d_out, int out_size, void* d_ws, size_t ws_size,
                              hipStream_t stream) {
  const float* in_key      = (const float*)d_in[0];
  const float* in_query    = (const float*)d_in[1];
  const float* dist        = (const float*)d_in[2];
  const float* W_key       = (const float*)d_in[3];
  const float* W_query     = (const float*)d_in[4];
  const float* W_bc        = (const float*)d_in[5];
  const float* W_dt        = (const float*)d_in[6];
  const float* dt_bias     = (const float*)d_in[7];
  const float* A_log       = (const float*)d_in[8];
  const float* Dp          = (const float*)d_in[9];
  const float* W_out_key   = (const float*)d_in[10];
  const float* W_out_query = (const float*)d_in[11];
  const float* key_norm_w  = (const float*)d_in[12];
  const float* q_norm_w    = (const float*)d_in[13];
  const float* q_norm_b    = (const float*)d_in[14];

  float* ws = (float*)d_ws;
  size_t o = 0;
  float* zx    = ws + o; o += (size_t)BSZ * LSEQ * DKEY;      // 4,235,264
  float* initf = ws + o; o += (size_t)BSZ * NQ * DI;          //   307,200
  float* dAw   = ws + o; o += (size_t)BSZ * LSEQ * NH * NQ;   // 9,830,400
  float* dtBw  = ws + o; o += (size_t)BSZ * LSEQ * NH * NQ;   // 9,830,400
  float* Cw    = ws + o; o += (size_t)BSZ * LSEQ * NQ;        // 1,228,800
  float* y_ws  = ws + o; o += (size_t)4 * LSEQ * DI;          // 4,194,304
  float* S_ws  = ws + o; o += (size_t)4 * NH * PD * NQ;       //   614,400
  float* kf    = ws + o; o += (size_t)BSZ * LSEQ * DI;        // 2,097,152
  float* lastn = ws + o; o += (size_t)BSZ * NQ * DI;          //   307,200

  float* out_key   = (float*)d_out;
  float* out_query = out_key + (size_t)BSZ * LSEQ * DM;

  dim3 blk(256);

  // zxbcdt = in_key @ W_key^T          [4096 x 1034], K=256
  gemm_nt_f32_wmma<<<dim3((DKEY + 255) / 256, (BSZ * LSEQ + 31) / 32), blk, 0, stream>>>(
      in_key, W_key, zx, BSZ * LSEQ, DKEY, DM, DKEY);
  // init = in_query @ W_query^T        [600 x 512], K=256
  gemm_nt_f32_wmma<<<dim3((DI + 255) / 256, (BSZ * NQ + 31) / 32), blk, 0, stream>>>(
      in_query, W_query, initf, BSZ * NQ, DI, DM, DI);

  // dist projections -> dA, dtB, C
  size_t blq = (size_t)BSZ * LSEQ * NQ;
  dist_proj_kernel<<<(unsigned)((blq + 255) / 256), blk, 0, stream>>>(
      dist, zx, W_bc, W_dt, dt_bias, A_log, dAw, dtBw, Cw);

  // bidirectional selective scan (32 independent (dir,b,h) scans)
  scan_kernel<<<32, blk, 0, stream>>>(zx, dAw, dtBw, Cw, initf, y_ws, S_ws);

  // combine + gate + RMSNorm
  gate_rms_kernel<<<BSZ * LSEQ, blk, 0, stream>>>(y_ws, zx, Dp, key_norm_w, kf);
  // final state combine + LayerNorm
  state_ln_kernel<<<BSZ * NQ, blk, 0, stream>>>(S_ws, q_norm_w, q_norm_b, lastn);

  // out_key = kf @ W_out_key^T         [4096 x 256], K=512
  gemm_nt_f32_wmma<<<dim3((DM + 255) / 256, (BSZ * LSEQ + 31) / 32), blk, 0, stream>>>(
      kf, W_out_key, out_key, BSZ * LSEQ, DM, DI, DM);
  // out_query = lastn @ W_out_query^T  [600 x 256], K=512
  gemm_nt_f32_wmma<<<dim3((DM + 255) / 256, (BSZ * NQ + 31) / 32), blk, 0, stream>>>(
      lastn, W_out_query, out_query, BSZ * NQ, DM, DI, DM);

  (void)in_sizes; (void)n_in; (void)out_size; (void)ws_size;
}